// gnn_conv_670014898743
// MI455X (gfx1250) — compile-verified
//
#include <hip/hip_runtime.h>

typedef __attribute__((ext_vector_type(2))) float v2f;
typedef __attribute__((ext_vector_type(8))) float v8f;

constexpr int NN = 12288;   // graph nodes
constexpr int FF = 32;      // feature width
constexpr int NT = 5;       // K+1 Chebyshev terms
constexpr int CHUNK  = 128;          // K-rows of Tprev staged per LDS buffer
constexpr int CS     = 132;          // LDS column stride (dwords): 128 + 4 pad -> conflict-free b64 reads
constexpr int NCHUNK = NN / CHUNK;   // 96

#if defined(__has_builtin)
#if __has_builtin(__builtin_amdgcn_global_load_async_to_lds_b32)
#define USE_ASYNC_LDS 1
#endif
#if __has_builtin(__builtin_amdgcn_s_wait_asynccnt)
#define HAVE_WAIT_ASYNC_BUILTIN 1
#endif
#endif

#if defined(USE_ASYNC_LDS)
#if defined(HAVE_WAIT_ASYNC_BUILTIN)
#define WAIT_ASYNC(n) __builtin_amdgcn_s_wait_asynccnt(n)
#else
#define WAIT_ASYNC(n) asm volatile("s_wait_asynccnt " #n ::: "memory")
#endif
#else
#define WAIT_ASYNC(n) ((void)0)
#endif

// Stage CHUNK x FF floats of Tprev into LDS, transposed to column-major
// (element (k,c) at dword c*CS + k). Global side: 128 lanes x 32 iters of
// perfectly coalesced b32; LDS side: per-lane scatter (async engine handles it).
__device__ static inline void stage_chunk(const float* __restrict__ Tprev, int k0,
                                          float* __restrict__ sbuf, int t)
{
    const float* g = Tprev + (size_t)k0 * FF + t;          // + i*128: coalesced stream
    float*       l = sbuf + (t & 31) * CS + (t >> 5);      // + i*4:   column scatter
#pragma unroll
    for (int i = 0; i < 32; ++i) {
#if defined(USE_ASYNC_LDS)
        __builtin_amdgcn_global_load_async_to_lds_b32(
            (__attribute__((address_space(1))) int*)(uintptr_t)(g + i * 128),
            (__attribute__((address_space(3))) int*)(unsigned)(uintptr_t)(l + i * 4),
            0, 0);
#else
        l[i * 4] = g[i * 128];
#endif
    }
}

// Tnext = alpha * (L @ Tprev) + beta * Tm2
// One wave owns a 16-row x 32-col strip (two 16x16 f32 WMMA accumulators).
// L streams from HBM exactly once per pass; Tprev chunks are double-buffered
// in LDS (column-major) so B fragments are single mov-free ds_load_b64 ops.
__global__ __launch_bounds__(128) void cheb_step_kernel(
    const float* __restrict__ L, const float* __restrict__ Tprev,
    const float* __restrict__ Tm2, float* __restrict__ Tnext,
    float alpha, float beta)
{
    __shared__ float sB[2][FF * CS];   // 2 x 16.5 KB

    const int t    = threadIdx.x;
    const int lane = t & 31;
    const int wv   = t >> 5;             // 0..3
    const int m    = lane & 15;          // M (A) / N (B,C,D) index
    const int r    = lane >> 4;          // half-wave K selector
    const int rbase = blockIdx.x * 64 + wv * 16;

    // A fragment: lane m holds row (rbase+m); v0/v1 hold K = 2r, 2r+1
    const float* __restrict__ Arow = L + (size_t)(rbase + m) * NN + 2 * r;

    v8f acc0 = {0.f,0.f,0.f,0.f,0.f,0.f,0.f,0.f};
    v8f acc1 = acc0;

    stage_chunk(Tprev, 0, sB[0], t);

    for (int c = 0; c < NCHUNK; ++c) {
        if (c + 1 < NCHUNK) {
            stage_chunk(Tprev, (c + 1) * CHUNK, sB[(c + 1) & 1], t);
            WAIT_ASYNC(32);  // async loads retire in order: <=32 left => chunk c landed
        } else {
            WAIT_ASYNC(0);
        }
        __syncthreads();

        const float* __restrict__ A0  = Arow + c * CHUNK;
        // Column-major B: fragment {K=kk+2r, kk+2r+1} at column col is contiguous.
        const float* __restrict__ sb0 = &sB[c & 1][m * CS + 2 * r];          // cols 0..15
        const float* __restrict__ sb1 = sb0 + 16 * CS;                        // cols 16..31
        __builtin_prefetch(A0 + 2 * CHUNK, 0, 3);   // pull L segment two chunks ahead toward WGP

#pragma unroll 8
        for (int kk = 0; kk < CHUNK; kk += 4) {
            v2f a  = *(const v2f*)(A0 + kk);         // global_load_b64 (HBM stream)
            v2f b0 = *(const v2f*)(sb0 + kk);        // ds_load_b64, imm offset, no banksplit
            v2f b1 = *(const v2f*)(sb1 + kk);
            acc0 = __builtin_amdgcn_wmma_f32_16x16x4_f32(
                false, a, false, b0, (short)0, acc0, false, false);
            acc1 = __builtin_amdgcn_wmma_f32_16x16x4_f32(
                false, a, false, b1, (short)0, acc1, false, false);
        }
        __syncthreads();
    }

    // C/D layout: VGPR v -> M = v + 8*r, N = m
#pragma unroll
    for (int v = 0; v < 8; ++v) {
        const int row = rbase + v + r * 8;
        const size_t o0 = (size_t)row * FF + m;
        Tnext[o0]      = alpha * acc0[v] + beta * Tm2[o0];
        Tnext[o0 + 16] = alpha * acc1[v] + beta * Tm2[o0 + 16];
    }
}

// y[n,o] = sum_k T_k @ W[:,:,k]   (W layout: W[i][o][k] = W[i*FF*NT + o*NT + k])
__global__ __launch_bounds__(128) void cheb_combine_kernel(
    const float* __restrict__ T0, const float* __restrict__ T1,
    const float* __restrict__ T2, const float* __restrict__ T3,
    const float* __restrict__ T4, const float* __restrict__ W,
    float* __restrict__ y)
{
    const int lane = threadIdx.x & 31;
    const int wv   = threadIdx.x >> 5;
    const int m    = lane & 15;
    const int r    = lane >> 4;
    const int rbase = blockIdx.x * 64 + wv * 16;

    const float* Ts[NT] = {T0, T1, T2, T3, T4};
    v8f acc0 = {0.f,0.f,0.f,0.f,0.f,0.f,0.f,0.f};
    v8f acc1 = acc0;

#pragma unroll
    for (int k = 0; k < NT; ++k) {
        const float* __restrict__ T = Ts[k];
        const float* __restrict__ Arow = T + (size_t)(rbase + m) * FF + 2 * r;
#pragma unroll
        for (int i0 = 0; i0 < FF; i0 += 4) {
            v2f a = *(const v2f*)(Arow + i0);
            const int i = i0 + 2 * r;
            v2f b0, b1;
            b0.x = W[(size_t)(i    ) * FF * NT + (m     ) * NT + k];
            b0.y = W[(size_t)(i + 1) * FF * NT + (m     ) * NT + k];
            b1.x = W[(size_t)(i    ) * FF * NT + (m + 16) * NT + k];
            b1.y = W[(size_t)(i + 1) * FF * NT + (m + 16) * NT + k];
            acc0 = __builtin_amdgcn_wmma_f32_16x16x4_f32(
                false, a, false, b0, (short)0, acc0, false, false);
            acc1 = __builtin_amdgcn_wmma_f32_16x16x4_f32(
                false, a, false, b1, (short)0, acc1, false, false);
        }
    }

#pragma unroll
    for (int v = 0; v < 8; ++v) {
        const int row = rbase + v + r * 8;
        const size_t o0 = (size_t)row * FF + m;
        y[o0]      = acc0[v];
        y[o0 + 16] = acc1[v];
    }
}

extern "C" void kernel_launch(void* const* d_in, const int* in_sizes, int n_in,
                              void* d_out, int out_size, void* d_ws, size_t ws_size,
                              hipStream_t stream) {
    (void)in_sizes; (void)n_in; (void)out_size; (void)ws_size;
    const float* x = (const float*)d_in[0];   // (12288, 32)
    const float* L = (const float*)d_in[1];   // (12288, 12288)
    const float* W = (const float*)d_in[2];   // (32, 32, 5)
    float* y = (float*)d_out;                 // (12288, 32)

    float* T1 = (float*)d_ws;                 // 4 x 1.5 MB scratch
    float* T2 = T1 + (size_t)NN * FF;
    float* T3 = T2 + (size_t)NN * FF;
    float* T4 = T3 + (size_t)NN * FF;

    dim3 grid(NN / 64), block(128);
    // T1 = L @ x
    cheb_step_kernel<<<grid, block, 0, stream>>>(L, x,  x,  T1, 1.0f,  0.0f);
    // T_k = 2 L T_{k-1} - T_{k-2}
    cheb_step_kernel<<<grid, block, 0, stream>>>(L, T1, x,  T2, 2.0f, -1.0f);
    cheb_step_kernel<<<grid, block, 0, stream>>>(L, T2, T1, T3, 2.0f, -1.0f);
    cheb_step_kernel<<<grid, block, 0, stream>>>(L, T3, T2, T4, 2.0f, -1.0f);
    // y = sum_k T_k @ W_k
    cheb_combine_kernel<<<grid, block, 0, stream>>>(x, T1, T2, T3, T4, W, y);
}